// EfficientFi_35862976921835
// MI455X (gfx1250) — compile-verified
//
#include <hip/hip_runtime.h>
#include <hip/hip_bf16.h>

typedef __attribute__((ext_vector_type(2))) float v2f;
typedef __attribute__((ext_vector_type(8))) float v8f;

__device__ __forceinline__ v8f wmma4(v2f a, v2f b, v8f c) {
  // D = A(16x4) * B(4x16) + C(16x16), fp32, wave32
  return __builtin_amdgcn_wmma_f32_16x16x4_f32(false, a, false, b, (short)0, c,
                                               false, false);
}

// ---------------- K1: conv1 3->48, stride2 pad1, ReLU.  x(512,3,80,80)->h1(512,48,40,40)
__global__ void k_conv1(const float* __restrict__ x, const float* __restrict__ w,
                        const float* __restrict__ bias, float* __restrict__ out) {
  int i = blockIdx.x * blockDim.x + threadIdx.x;
  const int total = 512 * 48 * 40 * 40;
  if (i >= total) return;
  int ox = i % 40, oy = (i / 40) % 40, co = (i / 1600) % 48, b = i / (48 * 1600);
  float v = bias[co];
#pragma unroll
  for (int ci = 0; ci < 3; ++ci)
#pragma unroll
    for (int kh = 0; kh < 3; ++kh) {
      int iy = oy * 2 - 1 + kh;
      if (iy < 0 || iy >= 80) continue;
#pragma unroll
      for (int kw = 0; kw < 3; ++kw) {
        int ix = ox * 2 - 1 + kw;
        if (ix < 0 || ix >= 80) continue;
        v += x[((b * 3 + ci) * 80 + iy) * 80 + ix] *
             w[((co * 3 + ci) * 3 + kh) * 3 + kw];
      }
    }
  out[i] = v > 0.f ? v : 0.f;
}

// ---------------- K2/K4: 2x2 maxpool with argmax index (dy*2+dx, first-max wins)
__global__ void k_maxpool(const float* __restrict__ in, float* __restrict__ out,
                          int* __restrict__ idx, int BC, int H, int W) {
  int h = H >> 1, w = W >> 1;
  int total = BC * h * w;
  int i = blockIdx.x * blockDim.x + threadIdx.x;
  if (i >= total) return;
  int xw = i % w, yh = (i / w) % h, bc = i / (w * h);
  const float* p = in + (bc * H + yh * 2) * W + xw * 2;
  float v0 = p[0], v1 = p[1], v2 = p[W], v3 = p[W + 1];
  float best = v0; int bi = 0;
  if (v1 > best) { best = v1; bi = 1; }
  if (v2 > best) { best = v2; bi = 2; }
  if (v3 > best) { best = v3; bi = 3; }
  out[i] = best; idx[i] = bi;
}

// ---------------- K3: conv2 48->128 stride2 pad1 ReLU via implicit-GEMM WMMA.
// M = 512*100 pixels (16/block), N = 128 (8 waves * 16), K = 9 chunks of 48.
__global__ void k_conv2_wmma(const float* __restrict__ p1, const float* __restrict__ w2,
                             const float* __restrict__ bias, float* __restrict__ h2) {
  __shared__ float As[16 * 48];
  const int tid = threadIdx.x;
  const int lane = tid & 31, wid = tid >> 5;
  const int row = lane & 15, hi = lane >> 4;
  const int m0 = blockIdx.x * 16;
  const int n = wid * 16 + row;
  v8f acc = {0.f, 0.f, 0.f, 0.f, 0.f, 0.f, 0.f, 0.f};
  for (int kh = 0; kh < 3; ++kh)
    for (int kw = 0; kw < 3; ++kw) {
      for (int e = tid; e < 16 * 48; e += 256) {
        int r = e / 48, ci = e % 48;
        int m = m0 + r, b = m / 100, p = m % 100, oy = p / 10, ox = p % 10;
        int iy = oy * 2 - 1 + kh, ix = ox * 2 - 1 + kw;
        float val = 0.f;
        if (iy >= 0 && iy < 20 && ix >= 0 && ix < 20)
          val = p1[((b * 48 + ci) * 20 + iy) * 20 + ix];
        As[e] = val;
      }
      __syncthreads();
#pragma unroll
      for (int s = 0; s < 12; ++s) {
        int k0 = s * 4 + hi * 2;
        v2f av, bv;
        av.x = As[row * 48 + k0]; av.y = As[row * 48 + k0 + 1];
        int widx = (n * 48 + k0) * 9 + kh * 3 + kw;
        bv.x = w2[widx]; bv.y = w2[widx + 9];
        acc = wmma4(av, bv, acc);
      }
      __syncthreads();
    }
  float bb = bias[n];
#pragma unroll
  for (int v = 0; v < 8; ++v) {
    int m = m0 + v + hi * 8;
    int b = m / 100, p = m % 100, oy = p / 10, ox = p % 10;
    float val = acc[v] + bb;
    h2[((b * 128 + n) * 10 + oy) * 10 + ox] = val > 0.f ? val : 0.f;
  }
}

// ---------------- K5: pre-VQ 1x1: z(B,128,5,5) -> z2d[12800][64] (WMMA, K=128)
__global__ void k_preproj_wmma(const float* __restrict__ z, const float* __restrict__ pw,
                               const float* __restrict__ pb, float* __restrict__ z2d) {
  __shared__ float As[16 * 128];
  const int tid = threadIdx.x;                  // 128 threads, 4 waves
  const int lane = tid & 31, wid = tid >> 5;
  const int row = lane & 15, hi = lane >> 4;
  const int m0 = blockIdx.x * 16;
  for (int e = tid; e < 16 * 128; e += 128) {
    int r = e >> 7, c = e & 127;
    int m = m0 + r, b = m / 25, p = m % 25;
    As[e] = z[(b * 128 + c) * 25 + p];
  }
  __syncthreads();
  v8f acc = {0.f, 0.f, 0.f, 0.f, 0.f, 0.f, 0.f, 0.f};
  const int n = wid * 16 + row;                 // n < 64
#pragma unroll 4
  for (int s = 0; s < 32; ++s) {
    int k0 = s * 4 + hi * 2;
    v2f av, bv;
    av.x = As[row * 128 + k0]; av.y = As[row * 128 + k0 + 1];
    bv.x = pw[n * 128 + k0];   bv.y = pw[n * 128 + k0 + 1];
    acc = wmma4(av, bv, acc);
  }
  float bb = pb[n];
#pragma unroll
  for (int v = 0; v < 8; ++v)
    z2d[(m0 + v + hi * 8) * 64 + n] = acc[v] + bb;
}

// ---------------- codebook squared norms
__global__ void k_cnorm(const float* __restrict__ cb, float* __restrict__ cn) {
  int k = blockIdx.x * blockDim.x + threadIdx.x;
  if (k >= 256) return;
  float s = 0.f;
#pragma unroll 8
  for (int d = 0; d < 64; ++d) { float v = cb[k * 64 + d]; s += v * v; }
  cn[k] = s;
}

// ---------------- cb_trans[256][128] = codebook @ trans_w^T (WMMA)
__global__ void k_cbtrans_wmma(const float* __restrict__ cb, const float* __restrict__ tw,
                               float* __restrict__ ct) {
  const int tid = threadIdx.x;                  // 256 threads, 8 waves
  const int lane = tid & 31, wid = tid >> 5;
  const int row = lane & 15, hi = lane >> 4;
  const int m0 = blockIdx.x * 16;
  const int n = wid * 16 + row;                 // n < 128
  v8f acc = {0.f, 0.f, 0.f, 0.f, 0.f, 0.f, 0.f, 0.f};
#pragma unroll
  for (int s = 0; s < 16; ++s) {
    int k0 = s * 4 + hi * 2;
    v2f av, bv;
    av.x = cb[(m0 + row) * 64 + k0]; av.y = cb[(m0 + row) * 64 + k0 + 1];
    bv.x = tw[n * 64 + k0];          bv.y = tw[n * 64 + k0 + 1];
    acc = wmma4(av, bv, acc);
  }
#pragma unroll
  for (int v = 0; v < 8; ++v)
    ct[(m0 + v + hi * 8) * 128 + n] = acc[v];
}

// ---------------- K6: scores = z2d @ cb^T (WMMA), dist = |c|^2 - 2*score,
// per-row argmin over 256; writes per-row loss partial (deterministic).
__global__ void k_vq_argmin(const float* __restrict__ z2d, const float* __restrict__ cb,
                            const float* __restrict__ cn, int* __restrict__ idx_out,
                            float* __restrict__ rowloss) {
  __shared__ float sc[16 * 256];
  __shared__ float rv[16 * 16];
  __shared__ int   ri[16 * 16];
  const int tid = threadIdx.x;                  // 256 threads, 8 waves * 32 cols
  const int lane = tid & 31, wid = tid >> 5;
  const int row = lane & 15, hi = lane >> 4;
  const int m0 = blockIdx.x * 16;
  const int nb = wid * 32;
  v8f acc0 = {0.f, 0.f, 0.f, 0.f, 0.f, 0.f, 0.f, 0.f};
  v8f acc1 = {0.f, 0.f, 0.f, 0.f, 0.f, 0.f, 0.f, 0.f};
  const int n0 = nb + row, n1 = nb + 16 + row;
#pragma unroll
  for (int s = 0; s < 16; ++s) {
    int k0 = s * 4 + hi * 2;
    v2f av, bv0, bv1;
    av.x = z2d[(m0 + row) * 64 + k0]; av.y = z2d[(m0 + row) * 64 + k0 + 1];
    bv0.x = cb[n0 * 64 + k0];         bv0.y = cb[n0 * 64 + k0 + 1];
    bv1.x = cb[n1 * 64 + k0];         bv1.y = cb[n1 * 64 + k0 + 1];
    acc0 = wmma4(av, bv0, acc0);
    acc1 = wmma4(av, bv1, acc1);
  }
  float c0 = cn[n0], c1 = cn[n1];
#pragma unroll
  for (int v = 0; v < 8; ++v) {
    int r = v + hi * 8;
    sc[r * 256 + n0] = c0 - 2.f * acc0[v];
    sc[r * 256 + n1] = c1 - 2.f * acc1[v];
  }
  __syncthreads();
  {  // stage 1: each thread scans a 16-wide segment of one row
    int r = tid >> 4, seg = tid & 15;
    const float* srow = sc + r * 256 + seg * 16;
    float best = srow[0]; int bi = seg * 16;
    for (int j = 1; j < 16; ++j)
      if (srow[j] < best) { best = srow[j]; bi = seg * 16 + j; }
    rv[r * 16 + seg] = best; ri[r * 16 + seg] = bi;
  }
  __syncthreads();
  if (tid < 16) {  // stage 2: combine (strict < keeps lowest index = first min)
    int r = tid;
    float best = rv[r * 16]; int bi = ri[r * 16];
    for (int seg = 1; seg < 16; ++seg)
      if (rv[r * 16 + seg] < best) { best = rv[r * 16 + seg]; bi = ri[r * 16 + seg]; }
    idx_out[m0 + r] = bi;
    float s = 0.f;
    for (int d = 0; d < 64; ++d) {
      float diff = cb[bi * 64 + d] - z2d[(m0 + r) * 64 + d];
      s += diff * diff;
    }
    rowloss[m0 + r] = s;   // fixed position, fixed order -> deterministic
  }
}

// ---------------- K7: latent(B,128,5,5) = cb_trans[idx] + trans_b (gather)
__global__ void k_latent(const int* __restrict__ idx, const float* __restrict__ ct,
                         const float* __restrict__ tb, float* __restrict__ lat) {
  int i = blockIdx.x * blockDim.x + threadIdx.x;
  if (i >= 512 * 128 * 25) return;
  int p = i % 25, c = (i / 25) % 128, b = i / (128 * 25);
  lat[i] = ct[idx[b * 25 + p] * 128 + c] + tb[c];
}

// ---------------- K8/K10: maxunpool 2x2 (thread per output, no memset needed)
__global__ void k_unpool(const float* __restrict__ val, const int* __restrict__ idx,
                         float* __restrict__ out, int BC, int h, int w) {
  int W2 = w * 2;
  int total = BC * h * w * 4;
  int i = blockIdx.x * blockDim.x + threadIdx.x;
  if (i >= total) return;
  int x = i % W2, y = (i / W2) % (h * 2), bc = i / (4 * h * w);
  int sel = (y & 1) * 2 + (x & 1);
  int in = (bc * h + (y >> 1)) * w + (x >> 1);
  out[i] = (idx[in] == sel) ? val[in] : 0.f;
}

// ---------------- K9: convT 128->48 stride2 pad1 outpad1 ReLU via WMMA.
// Gather form: out[oy] += in[(oy+1-kh)/2] * dw1[ci][co][kh][kw] when divisible.
// M = 512*400 (16/block), N = 48 (3 waves), K = 9 chunks of 128.
__global__ void k_dconv1_wmma(const float* __restrict__ u2, const float* __restrict__ dw,
                              const float* __restrict__ db, float* __restrict__ d1) {
  __shared__ float As[16 * 128];
  const int tid = threadIdx.x;                  // 96 threads, 3 waves
  const int lane = tid & 31, wid = tid >> 5;
  const int row = lane & 15, hi = lane >> 4;
  const int m0 = blockIdx.x * 16;
  const int n = wid * 16 + row;                 // out channel < 48
  v8f acc = {0.f, 0.f, 0.f, 0.f, 0.f, 0.f, 0.f, 0.f};
  for (int kh = 0; kh < 3; ++kh)
    for (int kw = 0; kw < 3; ++kw) {
      for (int e = tid; e < 16 * 128; e += 96) {
        int r = e >> 7, ci = e & 127;
        int m = m0 + r, b = m / 400, p = m % 400, oy = p / 20, ox = p % 20;
        int ty = oy + 1 - kh, tx = ox + 1 - kw;
        float vv = 0.f;
        if (ty >= 0 && tx >= 0 && !(ty & 1) && !(tx & 1)) {
          int iy = ty >> 1, ix = tx >> 1;
          if (iy < 10 && ix < 10) vv = u2[((b * 128 + ci) * 10 + iy) * 10 + ix];
        }
        As[e] = vv;
      }
      __syncthreads();
#pragma unroll 4
      for (int s = 0; s < 32; ++s) {
        int k0 = s * 4 + hi * 2;
        v2f av, bv;
        av.x = As[row * 128 + k0]; av.y = As[row * 128 + k0 + 1];
        int widx = (k0 * 48 + n) * 9 + kh * 3 + kw;
        bv.x = dw[widx]; bv.y = dw[widx + 432];   // next ci
        acc = wmma4(av, bv, acc);
      }
      __syncthreads();
    }
  float bb = db[n];
#pragma unroll
  for (int v = 0; v < 8; ++v) {
    int m = m0 + v + hi * 8;
    int b = m / 400, p = m % 400, oy = p / 20, ox = p % 20;
    float vv = acc[v] + bb;
    d1[((b * 48 + n) * 20 + oy) * 20 + ox] = vv > 0.f ? vv : 0.f;
  }
}

// ---------------- K11: convT 48->3 stride2 pad1 outpad1, direct (Cout=3)
__global__ void k_dconv2(const float* __restrict__ u1, const float* __restrict__ dw,
                         const float* __restrict__ db, float* __restrict__ rx) {
  int i = blockIdx.x * blockDim.x + threadIdx.x;
  const int total = 512 * 3 * 80 * 80;
  if (i >= total) return;
  int ox = i % 80, oy = (i / 80) % 80, co = (i / 6400) % 3, b = i / 19200;
  float v = db[co];
  for (int kh = 0; kh < 3; ++kh) {
    int ty = oy + 1 - kh;
    if (ty < 0 || (ty & 1)) continue;
    int iy = ty >> 1;
    if (iy >= 40) continue;
    for (int kw = 0; kw < 3; ++kw) {
      int tx = ox + 1 - kw;
      if (tx < 0 || (tx & 1)) continue;
      int ix = tx >> 1;
      if (ix >= 40) continue;
      const float* up = u1 + (b * 48 * 40 + iy) * 40 + ix;
      const float* wp = dw + co * 9 + kh * 3 + kw;
#pragma unroll 8
      for (int ci = 0; ci < 48; ++ci) v += up[ci * 1600] * wp[ci * 27];
    }
  }
  rx[i] = v;
}

// ---------------- K12: head fc1 (512,3200)@(3200,32)+ReLU, WMMA (latent is flat f)
__global__ void k_fc1_wmma(const float* __restrict__ f, const float* __restrict__ w,
                           const float* __restrict__ bias, float* __restrict__ hid) {
  const int tid = threadIdx.x;                  // 64 threads, 2 waves
  const int lane = tid & 31, wid = tid >> 5;
  const int row = lane & 15, hi = lane >> 4;
  const int m0 = blockIdx.x * 16;
  const int n = wid * 16 + row;                 // n < 32
  v8f acc = {0.f, 0.f, 0.f, 0.f, 0.f, 0.f, 0.f, 0.f};
#pragma unroll 4
  for (int s = 0; s < 800; ++s) {
    int k0 = s * 4 + hi * 2;
    v2f av, bv;
    av.x = f[(m0 + row) * 3200 + k0]; av.y = f[(m0 + row) * 3200 + k0 + 1];
    bv.x = w[n * 3200 + k0];          bv.y = w[n * 3200 + k0 + 1];
    acc = wmma4(av, bv, acc);
  }
  float bb = bias[n];
#pragma unroll
  for (int v = 0; v < 8; ++v) {
    float vv = acc[v] + bb;
    hid[(m0 + v + hi * 8) * 32 + n] = vv > 0.f ? vv : 0.f;
  }
}

// ---------------- K13: fc2 (512,32)@(32,34) -> kp
__global__ void k_fc2(const float* __restrict__ hid, const float* __restrict__ w,
                      const float* __restrict__ bias, float* __restrict__ kp) {
  int i = blockIdx.x * blockDim.x + threadIdx.x;
  if (i >= 512 * 34) return;
  int o = i % 34, b = i / 34;
  float v = bias[o];
#pragma unroll
  for (int j = 0; j < 32; ++j) v += hid[b * 32 + j] * w[o * 32 + j];
  kp[i] = v;
}

// ---------------- K14: deterministic tree reduction of 12800 row losses
__global__ void k_loss_reduce(const float* __restrict__ rowloss, float* __restrict__ out0) {
  __shared__ float red[256];
  const int tid = threadIdx.x;
  float s = 0.f;
  for (int i = tid; i < 12800; i += 256) s += rowloss[i];  // fixed per-thread order
  red[tid] = s;
  __syncthreads();
  for (int off = 128; off > 0; off >>= 1) {                 // fixed tree order
    if (tid < off) red[tid] += red[tid + off];
    __syncthreads();
  }
  if (tid == 0) out0[0] = red[0] * (1.25f / (12800.f * 64.f));
}

extern "C" void kernel_launch(void* const* d_in, const int* in_sizes, int n_in,
                              void* d_out, int out_size, void* d_ws, size_t ws_size,
                              hipStream_t stream) {
  const float* x        = (const float*)d_in[0];
  const float* w1       = (const float*)d_in[1];
  const float* b1       = (const float*)d_in[2];
  const float* w2       = (const float*)d_in[3];
  const float* b2       = (const float*)d_in[4];
  const float* pre_w    = (const float*)d_in[5];
  const float* pre_b    = (const float*)d_in[6];
  const float* codebook = (const float*)d_in[7];
  const float* trans_w  = (const float*)d_in[8];
  const float* trans_b  = (const float*)d_in[9];
  const float* dw1      = (const float*)d_in[10];
  const float* db1      = (const float*)d_in[11];
  const float* dw2      = (const float*)d_in[12];
  const float* db2      = (const float*)d_in[13];
  const float* hw1      = (const float*)d_in[14];
  const float* hb1      = (const float*)d_in[15];
  const float* hw2      = (const float*)d_in[16];
  const float* hb2      = (const float*)d_in[17];
  float* out = (float*)d_out;
  float* ws = (float*)d_ws;
  int*   wsi = (int*)d_ws;

  // workspace offsets in 4-byte elements (buffers reused across phases)
  const size_t OF_A    = 0;         // h1 then u1   (39,321,600)
  const size_t OF_IDX1 = 39321600;  // int          ( 9,830,400)
  const size_t OF_C    = 49152000;  // p1 then d1   ( 9,830,400)
  const size_t OF_D    = 58982400;  // h2 then u2   ( 6,553,600)
  const size_t OF_E    = 65536000;  // z            ( 1,638,400)
  const size_t OF_IDX2 = 67174400;  // int          ( 1,638,400)
  const size_t OF_G    = 68812800;  // z2d          (   819,200)
  const size_t OF_H    = 69632000;  // latent       ( 1,638,400)
  const size_t OF_IDXQ = 71270400;  // int          (    12,800)
  const size_t OF_CN   = 71283200;  // cnorm        (       256)
  const size_t OF_CT   = 71283456;  // cb_trans     (    32,768)
  const size_t OF_HID  = 71316224;  // hid          (    16,384)
  const size_t OF_RL   = 71332608;  // per-row loss (    12,800)

  k_conv1<<<153600, 256, 0, stream>>>(x, w1, b1, ws + OF_A);
  k_maxpool<<<38400, 256, 0, stream>>>(ws + OF_A, ws + OF_C, wsi + OF_IDX1, 24576, 40, 40);
  k_conv2_wmma<<<3200, 256, 0, stream>>>(ws + OF_C, w2, b2, ws + OF_D);
  k_maxpool<<<6400, 256, 0, stream>>>(ws + OF_D, ws + OF_E, wsi + OF_IDX2, 65536, 10, 10);
  k_preproj_wmma<<<800, 128, 0, stream>>>(ws + OF_E, pre_w, pre_b, ws + OF_G);
  k_cnorm<<<1, 256, 0, stream>>>(codebook, ws + OF_CN);
  k_cbtrans_wmma<<<16, 256, 0, stream>>>(codebook, trans_w, ws + OF_CT);
  k_vq_argmin<<<800, 256, 0, stream>>>(ws + OF_G, codebook, ws + OF_CN,
                                       wsi + OF_IDXQ, ws + OF_RL);
  k_latent<<<6400, 256, 0, stream>>>(wsi + OF_IDXQ, ws + OF_CT, trans_b, ws + OF_H);
  k_unpool<<<25600, 256, 0, stream>>>(ws + OF_H, wsi + OF_IDX2, ws + OF_D, 65536, 5, 5);
  k_dconv1_wmma<<<12800, 96, 0, stream>>>(ws + OF_D, dw1, db1, ws + OF_C);
  k_unpool<<<153600, 256, 0, stream>>>(ws + OF_C, wsi + OF_IDX1, ws + OF_A, 24576, 20, 20);
  k_dconv2<<<38400, 256, 0, stream>>>(ws + OF_A, dw2, db2, out + 1);
  k_fc1_wmma<<<32, 64, 0, stream>>>(ws + OF_H, hw1, hb1, ws + OF_HID);
  k_fc2<<<68, 256, 0, stream>>>(ws + OF_HID, hw2, hb2, out + 1 + 9830400);
  k_loss_reduce<<<1, 256, 0, stream>>>(ws + OF_RL, out);
}